// Pool_41953240547782
// MI455X (gfx1250) — compile-verified
//
#include <hip/hip_runtime.h>
#include <cstdint>
#include <cstddef>

#define N_ROWS 524288
#define DIM 256
#define NSEG 8192
#define KPOOL 512

typedef __attribute__((ext_vector_type(2))) float v2f;
typedef __attribute__((ext_vector_type(8))) float v8f;

__device__ __forceinline__ void wmma_f32_16x16x4(v8f& acc, v2f a, v2f b) {
    // D = A(16x4) * B(4x16) + C, all f32, wave32 striped operands.
    acc = __builtin_amdgcn_wmma_f32_16x16x4_f32(
        /*neg_a=*/false, a, /*neg_b=*/false, b,
        /*c_mod=*/(short)0, acc, /*reuse_a=*/false, /*reuse_b=*/false);
}

// ---------------------------------------------------------------------------
// Init segment accumulators: seg_sum = 0, seg_max bits = -inf
// ---------------------------------------------------------------------------
__global__ void init_seg_kernel(float* __restrict__ seg_sum,
                                int* __restrict__ seg_max_bits) {
    const int idx = blockIdx.x * blockDim.x + threadIdx.x;   // < NSEG*DIM
    seg_sum[idx] = 0.0f;
    seg_max_bits[idx] = (int)0xFF800000u;                    // -inf
}

// ---------------------------------------------------------------------------
// GEMM1 fused with segment scatter:  h = x @ W1^T + b1, then
// seg_sum[bi[n]] += h[n], seg_max[bi[n]] = max(...), h never materialized.
// Block = 256 threads (8 waves) -> 128 rows x 256 cols tile.
// W1^T staged in 256 KB dynamic LDS, paired-k layout for ds_load_b64.
// ---------------------------------------------------------------------------
__global__ __launch_bounds__(256, 1)
void gemm1_scatter(const float* __restrict__ x, const float* __restrict__ W1,
                   const float* __restrict__ b1, const int* __restrict__ bi,
                   float* __restrict__ seg_sum, int* __restrict__ seg_max_bits) {
    extern __shared__ float lds[];   // DIM*DIM floats: [(k>>1)*DIM + d]*2 + (k&1)
    const int tid = threadIdx.x;

    // Stage W1^T (coalesced global reads along k).
    for (int idx = tid; idx < DIM * DIM; idx += 256) {
        const int d = idx >> 8;       // W1 row  (output column)
        const int k = idx & 255;      // W1 col  (reduction index)
        lds[(((k >> 1) * DIM + d) << 1) | (k & 1)] = W1[(size_t)d * DIM + k];
    }
    __syncthreads();

    const int wave = tid >> 5;
    const int lane = tid & 31;
    const int lo   = lane & 15;
    const int hi   = lane >> 4;                  // half-wave select
    const int r0   = blockIdx.x * 128 + wave * 16;

    // 16 accumulator tiles of 16x16 (cols t*16..t*16+15), seeded with bias.
    v8f acc[16];
#pragma unroll
    for (int t = 0; t < 16; ++t) {
        const float bv = b1[t * 16 + lo];
        acc[t] = (v8f){bv, bv, bv, bv, bv, bv, bv, bv};
    }

    const float* xrow = x + (size_t)(r0 + lo) * DIM;
    __builtin_prefetch(xrow, 0, 3);

#pragma unroll 2
    for (int k0 = 0; k0 < DIM; k0 += 4) {
        // A fragment 16x4: v0 = K=k0 (lanes 0-15) / K=k0+2 (16-31), v1 = K+1.
        const v2f a = *(const v2f*)(xrow + k0 + 2 * hi);
        const float* lb = lds + (((k0 + 2 * hi) >> 1) * DIM + lo) * 2;
#pragma unroll
        for (int t = 0; t < 16; ++t) {
            const v2f b = *(const v2f*)(lb + t * 32);   // (k, n) & (k+1, n)
            wmma_f32_16x16x4(acc[t], a, b);
        }
    }

    // Scatter epilogue: rows of C tile j -> global row r0 + hi*8 + j.
    int segs[8];
#pragma unroll
    for (int j = 0; j < 8; ++j) segs[j] = bi[r0 + hi * 8 + j];

#pragma unroll
    for (int t = 0; t < 16; ++t) {
        const int col = t * 16 + lo;
#pragma unroll
        for (int j = 0; j < 8; ++j) {
            const float v = acc[t][j];
            const size_t o = (size_t)segs[j] * DIM + col;
            atomicAdd(seg_sum + o, v);
            // float max via ordered-int trick (sign-safe, init = -inf bits)
            if (v >= 0.0f)
                atomicMax(seg_max_bits + o, __float_as_int(v));
            else
                atomicMin((unsigned int*)(seg_max_bits + o), __float_as_uint(v));
        }
    }
}

// ---------------------------------------------------------------------------
// Per-segment counts via binary search on the sorted index (no atomics).
// ---------------------------------------------------------------------------
__global__ void counts_kernel(const int* __restrict__ bi, float* __restrict__ counts) {
    const int b = blockIdx.x * blockDim.x + threadIdx.x;
    if (b >= NSEG) return;
    int lo0 = 0, hi0 = N_ROWS;
    while (lo0 < hi0) { int m = (lo0 + hi0) >> 1; if (bi[m] < b) lo0 = m + 1; else hi0 = m; }
    int lo1 = lo0, hi1 = N_ROWS;
    while (lo1 < hi1) { int m = (lo1 + hi1) >> 1; if (bi[m] < b + 1) lo1 = m + 1; else hi1 = m; }
    counts[b] = (float)(lo1 - lo0);
}

// ---------------------------------------------------------------------------
// Build xpool[B][512] = concat(mean, max-with-(-inf -> 0))
// ---------------------------------------------------------------------------
__global__ void build_xpool(const float* __restrict__ seg_sum,
                            const int* __restrict__ seg_max_bits,
                            const float* __restrict__ counts,
                            float* __restrict__ xpool) {
    const int idx = blockIdx.x * blockDim.x + threadIdx.x;   // < NSEG*DIM
    const int b = idx >> 8;
    const int k = idx & 255;
    const float c = fmaxf(counts[b], 1.0f);
    float m = __int_as_float(seg_max_bits[idx]);
    if (!__builtin_isfinite(m)) m = 0.0f;
    xpool[(size_t)b * KPOOL + k]       = seg_sum[idx] / c;
    xpool[(size_t)b * KPOOL + 256 + k] = m;
}

// ---------------------------------------------------------------------------
// GEMM2: out_pool = xpool @ Wp^T + bp   (8192 x 512 x 256), WMMA f32,
// Wp^T staged in two 256-row LDS chunks.
// ---------------------------------------------------------------------------
__global__ __launch_bounds__(256, 1)
void gemm2_pool(const float* __restrict__ xpool, const float* __restrict__ Wp,
                const float* __restrict__ bp, float* __restrict__ out_pool) {
    extern __shared__ float lds[];
    const int tid  = threadIdx.x;
    const int wave = tid >> 5;
    const int lane = tid & 31;
    const int lo   = lane & 15;
    const int hi   = lane >> 4;
    const int r0   = blockIdx.x * 128 + wave * 16;

    v8f acc[16];
#pragma unroll
    for (int t = 0; t < 16; ++t) {
        const float bv = bp[t * 16 + lo];
        acc[t] = (v8f){bv, bv, bv, bv, bv, bv, bv, bv};
    }

    const float* arow = xpool + (size_t)(r0 + lo) * KPOOL;

    for (int c = 0; c < 2; ++c) {
        const int kbase = c * 256;
        for (int idx = tid; idx < DIM * 256; idx += 256) {
            const int d = idx >> 8;
            const int k = idx & 255;
            lds[(((k >> 1) * DIM + d) << 1) | (k & 1)] = Wp[(size_t)d * KPOOL + kbase + k];
        }
        __syncthreads();

#pragma unroll 2
        for (int k0 = 0; k0 < 256; k0 += 4) {
            const v2f a = *(const v2f*)(arow + kbase + k0 + 2 * hi);
            const float* lb = lds + (((k0 + 2 * hi) >> 1) * DIM + lo) * 2;
#pragma unroll
            for (int t = 0; t < 16; ++t) {
                const v2f b = *(const v2f*)(lb + t * 32);
                wmma_f32_16x16x4(acc[t], a, b);
            }
        }
        __syncthreads();
    }

#pragma unroll
    for (int t = 0; t < 16; ++t) {
        const int col = t * 16 + lo;
#pragma unroll
        for (int j = 0; j < 8; ++j)
            out_pool[(size_t)(r0 + hi * 8 + j) * DIM + col] = acc[t][j];
    }
}

// ---------------------------------------------------------------------------
// Gather: out[n] = out_pool[bi[n]], float4 vectorized (the 512 MB stream).
// ---------------------------------------------------------------------------
__global__ void gather_out(const int* __restrict__ bi,
                           const float* __restrict__ out_pool,
                           float* __restrict__ out) {
    const size_t i = (size_t)blockIdx.x * blockDim.x + threadIdx.x;  // < N*64
    const int n = (int)(i >> 6);
    const int c = (int)(i & 63);
    const float4* src = (const float4*)out_pool;
    float4* dst = (float4*)out;
    dst[(size_t)n * 64 + c] = src[(size_t)bi[n] * 64 + c];
}

// ---------------------------------------------------------------------------
extern "C" void kernel_launch(void* const* d_in, const int* in_sizes, int n_in,
                              void* d_out, int out_size, void* d_ws, size_t ws_size,
                              hipStream_t stream) {
    const float* x  = (const float*)d_in[0];
    const int*   bi = (const int*)d_in[1];
    const float* W1 = (const float*)d_in[2];
    const float* b1 = (const float*)d_in[3];
    const float* Wp = (const float*)d_in[4];
    const float* bp = (const float*)d_in[5];
    float* out = (float*)d_out;

    // Workspace layout (bytes):
    char* ws = (char*)d_ws;
    const size_t SEG_BYTES = (size_t)NSEG * DIM * sizeof(float);   // 8 MB
    float* seg_sum      = (float*)(ws);
    int*   seg_max_bits = (int*)  (ws + SEG_BYTES);
    float* counts       = (float*)(ws + 2 * SEG_BYTES);
    float* xpool        = (float*)(ws + 2 * SEG_BYTES + NSEG * sizeof(float));
    float* out_pool     = (float*)(ws + 2 * SEG_BYTES + NSEG * sizeof(float)
                                      + (size_t)NSEG * KPOOL * sizeof(float));

    const size_t ldsBytes = (size_t)DIM * DIM * sizeof(float);     // 256 KB

    init_seg_kernel<<<NSEG * DIM / 256, 256, 0, stream>>>(seg_sum, seg_max_bits);
    gemm1_scatter<<<N_ROWS / 128, 256, ldsBytes, stream>>>(x, W1, b1, bi,
                                                           seg_sum, seg_max_bits);
    counts_kernel<<<NSEG / 256, 256, 0, stream>>>(bi, counts);
    build_xpool<<<NSEG * DIM / 256, 256, 0, stream>>>(seg_sum, seg_max_bits,
                                                      counts, xpool);
    gemm2_pool<<<NSEG / 128, 256, ldsBytes, stream>>>(xpool, Wp, bp, out_pool);
    gather_out<<<(N_ROWS * 64) / 256, 256, 0, stream>>>(bi, out_pool, out);
}